// MVGRLModelSimple_9491877724926
// MI455X (gfx1250) — compile-verified
//
#include <hip/hip_runtime.h>

#define NNODES 50000
#define NEDGES 1600000
#define DIM    128

typedef __attribute__((ext_vector_type(2))) float v2f;
typedef __attribute__((ext_vector_type(8))) float v8f;

// ---------------------------------------------------------------------------
// Dense H = X (N x 128) @ W (128 x 128), fp32, via V_WMMA_F32_16X16X4_F32.
// One block = 16 rows of X; 8 waves, one 16-col tile each.
// ---------------------------------------------------------------------------
__global__ __launch_bounds__(256)
void gemm_xw_wmma(const float* __restrict__ X, const float* __restrict__ W,
                  float* __restrict__ H)
{
  __shared__ float tileA[16][132];   // pitch 132 avoids 16-way bank conflicts
  const int t    = threadIdx.x;
  const int row0 = blockIdx.x * 16;

  // stage the 16x128 X tile: each thread copies 8 consecutive floats
  {
    int r  = t >> 4;          // 0..15
    int c0 = (t & 15) * 8;    // 0..120
    const float* src = X + (size_t)(row0 + r) * DIM + c0;
#pragma unroll
    for (int j = 0; j < 8; ++j) tileA[r][c0 + j] = src[j];
  }
  __syncthreads();

  const int wave = t >> 5;          // 0..7 -> column tile
  const int lane = t & 31;
  const int col0 = wave * 16;
  const int m    = lane & 15;       // row (A) / col (B) within tile
  const int half = lane >> 4;       // K sub-slot selector

  v8f c = {};
#pragma unroll 4
  for (int k = 0; k < DIM; k += 4) {
    const int kb = k + half * 2;
    v2f a, b;
    // A (16x4): lanes 0-15 hold {K,K+1} of row m; lanes 16-31 hold {K+2,K+3}
    a.x = tileA[m][kb];
    a.y = tileA[m][kb + 1];
    // B (4x16): lanes 0-15 hold {K,K+1} of col m; lanes 16-31 hold {K+2,K+3}
    b.x = W[(size_t)kb * DIM + col0 + m];
    b.y = W[(size_t)(kb + 1) * DIM + col0 + m];
    c = __builtin_amdgcn_wmma_f32_16x16x4_f32(false, a, false, b,
                                              (short)0, c, false, false);
  }

  // D layout: VGPR v -> row (v + 8*half), col (lane%16)
  float* out = H + (size_t)row0 * DIM + col0;
#pragma unroll
  for (int v = 0; v < 8; ++v)
    out[(size_t)(v + 8 * half) * DIM + m] = c[v];
}

// ---------------------------------------------------------------------------
// Degree / normalization helpers
// ---------------------------------------------------------------------------
__global__ void init_deg(float* __restrict__ deg_r, float* __restrict__ deg_d,
                         float* __restrict__ colsum_r, float* __restrict__ colsum_d,
                         int n)
{
  int i = blockIdx.x * blockDim.x + threadIdx.x;
  if (i < n)   { deg_r[i] = 1.0f; deg_d[i] = 1.0f; }  // self-loop weight
  if (i < DIM) { colsum_r[i] = 0.0f; colsum_d[i] = 0.0f; }
}

__global__ void accum_deg(const int* __restrict__ col, const float* __restrict__ ew,
                          float* __restrict__ deg, int ne)
{
  int e = blockIdx.x * blockDim.x + threadIdx.x;
  if (e < ne) atomicAdd(&deg[col[e]], ew ? ew[e] : 1.0f);
}

__global__ void make_dinv(const float* __restrict__ deg, float* __restrict__ dinv, int n)
{
  int i = blockIdx.x * blockDim.x + threadIdx.x;
  if (i < n) { float d = deg[i]; dinv[i] = d > 0.0f ? rsqrtf(d) : 0.0f; }
}

// ---------------------------------------------------------------------------
// Fused edge aggregation for one graph: computes BOTH the plain and the
// corrupted scatter in a single pass so per-edge index/weight/dinv loads and
// the norm computation are paid once.
//   out_plain[col[e]] += norm * H[row[e]]
//   out_corr [col[e]] += norm * H[perm[row[e]]]
// One wave per edge; lane handles 4 consecutive feature dims (float4).
// ---------------------------------------------------------------------------
__global__ __launch_bounds__(256)
void aggregate_fused(const float* __restrict__ H, const int* __restrict__ perm,
                     const int* __restrict__ row, const int* __restrict__ col,
                     const float* __restrict__ ew, const float* __restrict__ dinv,
                     float* __restrict__ out_plain, float* __restrict__ out_corr,
                     int ne)
{
  int e = blockIdx.x * 8 + (threadIdx.x >> 5);
  if (e >= ne) return;
  int lane = threadIdx.x & 31;

  int r = row[e], c = col[e];
  float w    = ew ? ew[e] : 1.0f;
  float norm = dinv[r] * w * dinv[c];
  int rp     = perm[r];

  float4 h0 = ((const float4*)(H + (size_t)r  * DIM))[lane];
  float4 h1 = ((const float4*)(H + (size_t)rp * DIM))[lane];

  size_t off = (size_t)c * DIM + lane * 4;
  float* o0 = out_plain + off;
  float* o1 = out_corr  + off;
  atomicAdd(o0 + 0, norm * h0.x);
  atomicAdd(o0 + 1, norm * h0.y);
  atomicAdd(o0 + 2, norm * h0.z);
  atomicAdd(o0 + 3, norm * h0.w);
  atomicAdd(o1 + 0, norm * h1.x);
  atomicAdd(o1 + 1, norm * h1.y);
  atomicAdd(o1 + 2, norm * h1.z);
  atomicAdd(o1 + 3, norm * h1.w);
}

// ---------------------------------------------------------------------------
// Finalize: h = prelu(agg + dinv^2 * H[perm(i)] + bias).  Optionally
// accumulates per-dim column sums (for the readout mean) with an LDS
// reduction so only 128 atomics leave each block.  Block = 16 nodes.
// ---------------------------------------------------------------------------
__global__ __launch_bounds__(256)
void finalize(float* __restrict__ agg, const float* __restrict__ H,
              const int* __restrict__ perm, const float* __restrict__ dinv,
              const float* __restrict__ bias, const float* __restrict__ a_ptr,
              float* __restrict__ colsum, int n)
{
  __shared__ float part[256];
  const int t    = threadIdx.x;
  const int d    = t & (DIM - 1);
  const int hn   = t >> 7;            // 0/1: which half of the 16 nodes
  const int base = blockIdx.x * 16;
  const float a  = a_ptr[0];
  const float b  = bias[d];

  float acc = 0.0f;
#pragma unroll
  for (int j = 0; j < 8; ++j) {
    int i = base + hn * 8 + j;
    if (i >= n) break;
    int src = perm ? perm[i] : i;
    size_t idx = (size_t)i * DIM + d;
    float di  = dinv[i];
    float val = agg[idx] + di * di * H[(size_t)src * DIM + d] + b;
    float h   = val >= 0.0f ? val : a * val;
    agg[idx] = h;
    acc += h;
  }
  if (colsum) {
    part[t] = acc;
    __syncthreads();
    if (t < DIM) atomicAdd(&colsum[t], part[t] + part[t + DIM]);
  }
}

// ---------------------------------------------------------------------------
// r = sigmoid(colsum/N); out_rsum = r1+r2; v = W_bil @ r.  One block, 128 thr.
// ---------------------------------------------------------------------------
__global__ void compute_rv(const float* __restrict__ colsum_r,
                           const float* __restrict__ colsum_d,
                           const float* __restrict__ Wb,
                           float* __restrict__ v1, float* __restrict__ v2,
                           float* __restrict__ out_rsum, int n)
{
  __shared__ float r1[DIM], r2[DIM];
  int d = threadIdx.x;
  float m1 = colsum_r[d] / (float)n;
  float m2 = colsum_d[d] / (float)n;
  float s1 = 1.0f / (1.0f + expf(-m1));
  float s2 = 1.0f / (1.0f + expf(-m2));
  r1[d] = s1; r2[d] = s2;
  out_rsum[d] = s1 + s2;
  __syncthreads();
  float a1 = 0.0f, a2 = 0.0f;
  const float* wrow = Wb + (size_t)d * DIM;
#pragma unroll 8
  for (int e = 0; e < DIM; ++e) { float w = wrow[e]; a1 += w * r1[e]; a2 += w * r2[e]; }
  v1[d] = a1; v2[d] = a2;
}

// ---------------------------------------------------------------------------
// Discriminator: out[g] = h_sec[i] . v_sec + bb.  One wave per output.
// ---------------------------------------------------------------------------
__global__ __launch_bounds__(256)
void disc_kernel(const float* __restrict__ h_diff, const float* __restrict__ h_real,
                 const float* __restrict__ h_diff_c, const float* __restrict__ h_real_c,
                 const float* __restrict__ v1, const float* __restrict__ v2,
                 const float* __restrict__ bb, float* __restrict__ out, int n)
{
  int g = blockIdx.x * 8 + (threadIdx.x >> 5);
  if (g >= 4 * n) return;
  int lane = threadIdx.x & 31;
  int sec  = g / n;
  int i    = g - sec * n;

  const float* h; const float* v;
  if (sec == 0)      { h = h_diff;   v = v1; }
  else if (sec == 1) { h = h_real;   v = v2; }
  else if (sec == 2) { h = h_diff_c; v = v1; }
  else               { h = h_real_c; v = v2; }

  float4 hv = ((const float4*)(h + (size_t)i * DIM))[lane];
  float4 vv = ((const float4*)v)[lane];
  float p = hv.x * vv.x + hv.y * vv.y + hv.z * vv.z + hv.w * vv.w;
#pragma unroll
  for (int off = 16; off > 0; off >>= 1) p += __shfl_down(p, off, 32);
  if (lane == 0) out[g] = p + bb[0];
}

__global__ void hsum_kernel(const float* __restrict__ a, const float* __restrict__ b,
                            float* __restrict__ out, size_t total)
{
  size_t i = (size_t)blockIdx.x * blockDim.x + threadIdx.x;
  if (i < total) out[i] = a[i] + b[i];
}

// ---------------------------------------------------------------------------
extern "C" void kernel_launch(void* const* d_in, const int* in_sizes, int n_in,
                              void* d_out, int out_size, void* d_ws, size_t ws_size,
                              hipStream_t stream)
{
  (void)in_sizes; (void)n_in; (void)out_size; (void)ws_size;

  const float* x      = (const float*)d_in[0];
  const int*   ei     = (const int*)  d_in[1];
  const int*   dei    = (const int*)  d_in[2];
  const float* dew    = (const float*)d_in[3];
  const int*   cidx   = (const int*)  d_in[4];
  const float* W_real = (const float*)d_in[5];
  const float* b_real = (const float*)d_in[6];
  const float* a_real = (const float*)d_in[7];
  const float* W_diff = (const float*)d_in[8];
  const float* b_diff = (const float*)d_in[9];
  const float* a_diff = (const float*)d_in[10];
  const float* W_bil  = (const float*)d_in[11];
  const float* b_bil  = (const float*)d_in[12];

  const int N = NNODES, E = NEDGES;
  const int* row  = ei;       const int* colr = ei  + E;
  const int* drow = dei;      const int* dcol = dei + E;

  float* ws = (float*)d_ws;
  const size_t nd = (size_t)N * DIM;
  float* H_real   = ws; ws += nd;
  float* H_diff   = ws; ws += nd;
  float* h_real   = ws; ws += nd;   // 4 aggregation buffers are contiguous
  float* h_diff   = ws; ws += nd;
  float* h_real_c = ws; ws += nd;
  float* h_diff_c = ws; ws += nd;
  float* deg_r    = ws; ws += N;
  float* deg_d    = ws; ws += N;
  float* dinv_r   = ws; ws += N;
  float* dinv_d   = ws; ws += N;
  float* colsum_r = ws; ws += DIM;
  float* colsum_d = ws; ws += DIM;
  float* v1       = ws; ws += DIM;
  float* v2       = ws; ws += DIM;

  float* out      = (float*)d_out;
  float* out_disc = out;
  float* out_rsum = out + (size_t)4 * N;
  float* out_hsum = out_rsum + DIM;

  // zero the 4 contiguous aggregation accumulators (graph-capture safe)
  hipMemsetAsync(h_real, 0, 4 * nd * sizeof(float), stream);

  init_deg<<<(N + 255) / 256, 256, 0, stream>>>(deg_r, deg_d, colsum_r, colsum_d, N);
  accum_deg<<<(E + 255) / 256, 256, 0, stream>>>(colr, nullptr, deg_r, E);
  accum_deg<<<(E + 255) / 256, 256, 0, stream>>>(dcol, dew,     deg_d, E);
  make_dinv<<<(N + 255) / 256, 256, 0, stream>>>(deg_r, dinv_r, N);
  make_dinv<<<(N + 255) / 256, 256, 0, stream>>>(deg_d, dinv_d, N);

  gemm_xw_wmma<<<N / 16, 256, 0, stream>>>(x, W_real, H_real);
  gemm_xw_wmma<<<N / 16, 256, 0, stream>>>(x, W_diff, H_diff);

  const int aggGrid = (E + 7) / 8;
  aggregate_fused<<<aggGrid, 256, 0, stream>>>(H_real, cidx, row,  colr, nullptr,
                                               dinv_r, h_real, h_real_c, E);
  aggregate_fused<<<aggGrid, 256, 0, stream>>>(H_diff, cidx, drow, dcol, dew,
                                               dinv_d, h_diff, h_diff_c, E);

  const int finGrid = (N + 15) / 16;
  finalize<<<finGrid, 256, 0, stream>>>(h_real,   H_real, nullptr, dinv_r, b_real, a_real, colsum_r, N);
  finalize<<<finGrid, 256, 0, stream>>>(h_diff,   H_diff, nullptr, dinv_d, b_diff, a_diff, colsum_d, N);
  finalize<<<finGrid, 256, 0, stream>>>(h_real_c, H_real, cidx,    dinv_r, b_real, a_real, nullptr,  N);
  finalize<<<finGrid, 256, 0, stream>>>(h_diff_c, H_diff, cidx,    dinv_d, b_diff, a_diff, nullptr,  N);

  compute_rv<<<1, DIM, 0, stream>>>(colsum_r, colsum_d, W_bil, v1, v2, out_rsum, N);

  disc_kernel<<<(4 * N + 7) / 8, 256, 0, stream>>>(h_diff, h_real, h_diff_c, h_real_c,
                                                   v1, v2, b_bil, out_disc, N);

  hsum_kernel<<<(int)((nd + 255) / 256), 256, 0, stream>>>(h_real, h_diff, out_hsum, nd);
}